// GINNet_32908039422341
// MI455X (gfx1250) — compile-verified
//
#include <hip/hip_runtime.h>

typedef __attribute__((ext_vector_type(2))) float v2f;
typedef __attribute__((ext_vector_type(8))) float v8f;

#define N_NODES 50000
#define N_EDGES 600000
#define K_DIM   128
#define BN_EPS  1e-5f

// ---------------------------------------------------------------------------
// agg = x  (self term of GIN, eps=0), vectorized float4 copy
// ---------------------------------------------------------------------------
__global__ void gin_copy_kernel(const float* __restrict__ src,
                                float* __restrict__ dst, int n4) {
    int i = blockIdx.x * blockDim.x + threadIdx.x;
    if (i < n4) {
        reinterpret_cast<float4*>(dst)[i] =
            reinterpret_cast<const float4*>(src)[i];
    }
}

// ---------------------------------------------------------------------------
// Zero-padded weight transpose: WT[n][k] = (n < Hout) ? W[k][n] : 0
// Lets the WMMA B-fragment pair (consecutive K at fixed n) be one b64 load
// and removes all bounds checks from the GEMM inner loop.
// ---------------------------------------------------------------------------
__global__ void gin_transpose_pad(const float* __restrict__ W,
                                  float* __restrict__ WT,
                                  int Hout, int totalPad) {   // totalPad = HoutPad*128
    int i = blockIdx.x * blockDim.x + threadIdx.x;
    if (i >= totalPad) return;
    int n = i >> 7;          // padded output-column index
    int k = i & 127;
    WT[i] = (n < Hout) ? W[k * Hout + n] : 0.0f;
}

// ---------------------------------------------------------------------------
// agg[dst] += x[src] over all edges. One thread per (edge, 4 features):
// one coalesced b128 row read + 4 global_atomic_add_f32 (L2-resident).
// ---------------------------------------------------------------------------
__global__ void gin_scatter_kernel(const float* __restrict__ x,
                                   const int* __restrict__ src_idx,
                                   const int* __restrict__ dst_idx,
                                   float* __restrict__ agg,
                                   int total) {               // = E * 32
    int i = blockIdx.x * blockDim.x + threadIdx.x;
    if (i >= total) return;
    int e  = i >> 5;
    int f4 = (i & 31) << 2;
    int s = src_idx[e];
    int d = dst_idx[e];
    const float4 v = *reinterpret_cast<const float4*>(x + (size_t)s * K_DIM + f4);
    float* p = agg + (size_t)d * K_DIM + f4;
    atomicAdd(p + 0, v.x);
    atomicAdd(p + 1, v.y);
    atomicAdd(p + 2, v.z);
    atomicAdd(p + 3, v.w);
}

// ---------------------------------------------------------------------------
// Fused GEMM + epilogue on V_WMMA_F32_16X16X4_F32 (full f32 precision).
//   out[N, HOUT] = epilogue( A[N,128] @ W[128,HOUT] + bias )
// W supplied pre-transposed & zero-padded: WT[n][k], n < HOUT_PAD.
// MODE 0: +bias | MODE 1: relu(+bias) | MODE 2: relu(bn(+bias))
//
// Register blocking: each wave owns one 16-row tile x CG=4 col tiles
// (16x64 output, 4 accumulators). Per K-step: 1 A b64 + 4 B b64 -> 4 WMMA.
// Fragment layout (ISA 7.12.2, wave32):
//   A 16x4 f32 : row M = lane%16; VGPR0/1 hold K = 2*(lane/16)+{0,1}
//   B  4x16 f32: col N = lane%16; VGPR0/1 hold K = 2*(lane/16)+{0,1}
//   C/D 16x16  : VGPR r holds M = r + 8*(lane/16), N = lane%16
// ---------------------------------------------------------------------------
template<int MODE, int HOUT>
__global__ void gin_gemm_wmma(const float* __restrict__ A,
                              const float* __restrict__ WT,
                              const float* __restrict__ bias,
                              const float* __restrict__ bn_g,
                              const float* __restrict__ bn_b,
                              const float* __restrict__ bn_m,
                              const float* __restrict__ bn_v,
                              float* __restrict__ out) {
    constexpr int CG         = 4;                       // col tiles per wave
    constexpr int COL_TILES  = (HOUT + 15) / 16;
    constexpr int COL_GROUPS = (COL_TILES + CG - 1) / CG;
    constexpr int ROW_TILES  = N_NODES / 16;            // 50000 % 16 == 0
    constexpr int TOTAL      = ROW_TILES * COL_GROUPS;

    const int wave = threadIdx.x >> 5;
    const int tile = blockIdx.x * (blockDim.x >> 5) + wave;
    if (tile >= TOTAL) return;                          // wave-uniform exit

    const int rowTile = tile / COL_GROUPS;
    const int colBase = (tile % COL_GROUPS) * (CG * 16);

    const int lane = threadIdx.x & 31;
    const int m    = lane & 15;                         // M (A) / N (B,C,D)
    const int hi   = lane >> 4;                         // K-half selector

    const float* Arow  = A  + (long long)(rowTile * 16 + m) * K_DIM + 2 * hi;
    const float* Wlane = WT + (long long)(colBase + m) * K_DIM + 2 * hi;

    v8f acc[CG] = {};
    #pragma unroll 4
    for (int k0 = 0; k0 < K_DIM; k0 += 4) {
        v2f a = *reinterpret_cast<const v2f*>(Arow + k0);   // one b64
        #pragma unroll
        for (int t = 0; t < CG; ++t) {
            v2f b = *reinterpret_cast<const v2f*>(Wlane + (t << 11) + k0); // one b64
            // (neg_a, A, neg_b, B, c_mod, C, reuse_a, reuse_b)
            acc[t] = __builtin_amdgcn_wmma_f32_16x16x4_f32(
                false, a, false, b, (short)0, acc[t], false, false);
        }
    }

    // Epilogue: each lane owns column n_t per tile, rows r + 8*hi.
    #pragma unroll
    for (int t = 0; t < CG; ++t) {
        const int  n  = colBase + t * 16 + m;
        const bool ok = (HOUT % (CG * 16) == 0) || (n < HOUT);
        float bias_v = ok ? bias[n] : 0.0f;
        float scale = 1.0f, shift = 0.0f;
        if constexpr (MODE == 2) {                      // only HOUT==128: n valid
            float rs = rsqrtf(bn_v[n] + BN_EPS);
            scale = bn_g[n] * rs;
            shift = bn_b[n] - bn_m[n] * scale;
        }
        float* orow = out + (long long)(rowTile * 16) * HOUT + (ok ? n : 0);
        #pragma unroll
        for (int r = 0; r < 8; ++r) {
            float val = acc[t][r] + bias_v;
            if constexpr (MODE == 2) val = val * scale + shift;
            if constexpr (MODE >= 1) val = fmaxf(val, 0.0f);
            if (ok) orow[(long long)(r + 8 * hi) * HOUT] = val;
        }
    }
}

// ---------------------------------------------------------------------------
// Orchestration
// ---------------------------------------------------------------------------
extern "C" void kernel_launch(void* const* d_in, const int* in_sizes, int n_in,
                              void* d_out, int out_size, void* d_ws, size_t ws_size,
                              hipStream_t stream) {
    (void)in_sizes; (void)n_in; (void)out_size; (void)ws_size;

    const float* x      = (const float*)d_in[0];
    const int*   edge   = (const int*)  d_in[1];   // [2, E]: src row then dst row
    const float* lin1_w = (const float*)d_in[2];
    const float* lin1_b = (const float*)d_in[3];
    const float* bn1_g  = (const float*)d_in[4];
    const float* bn1_b  = (const float*)d_in[5];
    const float* bn1_m  = (const float*)d_in[6];
    const float* bn1_v  = (const float*)d_in[7];
    const float* lin2_w = (const float*)d_in[8];
    const float* lin2_b = (const float*)d_in[9];
    const float* lin3_w = (const float*)d_in[10];
    const float* lin3_b = (const float*)d_in[11];
    const float* bn2_g  = (const float*)d_in[12];
    const float* bn2_b  = (const float*)d_in[13];
    const float* bn2_m  = (const float*)d_in[14];
    const float* bn2_v  = (const float*)d_in[15];
    const float* lin4_w = (const float*)d_in[16];
    const float* lin4_b = (const float*)d_in[17];
    const float* lin5_w = (const float*)d_in[18];
    const float* lin5_b = (const float*)d_in[19];
    const float* lin6_w = (const float*)d_in[20];
    const float* lin6_b = (const float*)d_in[21];

    const int* e_src = edge;
    const int* e_dst = edge + N_EDGES;

    // Workspace layout: 2 ping-pong activation buffers + 6 transposed weights
    const size_t NH = (size_t)N_NODES * K_DIM;
    float* B0  = (float*)d_ws;
    float* B1  = B0  + NH;
    float* WT1 = B1  + NH;
    float* WT2 = WT1 + 128 * 128;
    float* WT3 = WT2 + 128 * 128;
    float* WT4 = WT3 + 128 * 128;
    float* WT5 = WT4 + 128 * 128;
    float* WT6 = WT5 + 128 * 128;                  // 64 x 128 (zero-padded 40->64)

    // ---- Pre-transpose all weights (tiny; independent of activations) ----
    gin_transpose_pad<<<64, 256, 0, stream>>>(lin1_w, WT1, 128, 128 * 128);
    gin_transpose_pad<<<64, 256, 0, stream>>>(lin2_w, WT2, 128, 128 * 128);
    gin_transpose_pad<<<64, 256, 0, stream>>>(lin3_w, WT3, 128, 128 * 128);
    gin_transpose_pad<<<64, 256, 0, stream>>>(lin4_w, WT4, 128, 128 * 128);
    gin_transpose_pad<<<64, 256, 0, stream>>>(lin5_w, WT5, 128, 128 * 128);
    gin_transpose_pad<<<32, 256, 0, stream>>>(lin6_w, WT6,  40,  64 * 128);

    const int NH4      = (int)(NH / 4);
    const int cpBlocks = (NH4 + 255) / 256;
    const int scTotal  = N_EDGES * 32;             // thread per (edge, 4 feats)
    const int scBlocks = (scTotal + 255) / 256;
    const int g128     = ((N_NODES / 16) * 2 + 7) / 8;   // 2 col-groups of 64
    const int g40      = ((N_NODES / 16) * 1 + 7) / 8;   // 1 col-group

    // ---- GIN layer 1: agg = x + sum_{j->i} x_j ----
    gin_copy_kernel   <<<cpBlocks, 256, 0, stream>>>(x, B0, NH4);
    gin_scatter_kernel<<<scBlocks, 256, 0, stream>>>(x, e_src, e_dst, B0, scTotal);
    gin_gemm_wmma<2,128><<<g128, 256, 0, stream>>>(B0, WT1, lin1_b,
                                                   bn1_g, bn1_b, bn1_m, bn1_v, B1);
    gin_gemm_wmma<1,128><<<g128, 256, 0, stream>>>(B1, WT2, lin2_b,
                                                   nullptr, nullptr, nullptr, nullptr, B0);

    // ---- GIN layer 2 ----
    gin_copy_kernel   <<<cpBlocks, 256, 0, stream>>>(B0, B1, NH4);
    gin_scatter_kernel<<<scBlocks, 256, 0, stream>>>(B0, e_src, e_dst, B1, scTotal);
    gin_gemm_wmma<2,128><<<g128, 256, 0, stream>>>(B1, WT3, lin3_b,
                                                   bn2_g, bn2_b, bn2_m, bn2_v, B0);
    gin_gemm_wmma<1,128><<<g128, 256, 0, stream>>>(B0, WT4, lin4_b,
                                                   nullptr, nullptr, nullptr, nullptr, B1);

    // ---- Head MLP ----
    gin_gemm_wmma<1,128><<<g128, 256, 0, stream>>>(B1, WT5, lin5_b,
                                                   nullptr, nullptr, nullptr, nullptr, B0);
    gin_gemm_wmma<0,40> <<<g40,  256, 0, stream>>>(B0, WT6, lin6_b,
                                                   nullptr, nullptr, nullptr, nullptr,
                                                   (float*)d_out);
}